// PropellerNoiseGen_64304250356299
// MI455X (gfx1250) — compile-verified
//
#include <hip/hip_runtime.h>
#include <math.h>

// ---------------------------------------------------------------------------
// PropellerNoiseGen for MI455X (gfx1250, wave32)
//   out[b,t] = sum_k exp(log_A[k]) * sin(k*phi[b,t] + 2*phase_shift[b]*k)
//   phi = 2*pi * cumsum(speed_rps * 2 / 44100)
//
// v3: shuffle-free WMMA mapping (lanes l and l+16 share a time sample but
// carry odd/even harmonic parities, stepping by a 2*phi rotation), plus
// direct v_sin_f32/v_cos_f32 on fractional revolutions (the HW trig unit's
// native domain), eliminating the 2pi * (1/2pi) multiply pair.
// ---------------------------------------------------------------------------

typedef __attribute__((ext_vector_type(2))) float v2f;
typedef __attribute__((ext_vector_type(8))) float v8f;

constexpr int   T_LEN  = 441000;
constexpr int   NB     = 8;
constexpr int   NH     = 50;          // harmonics
constexpr int   NPAIR  = NH / 2;      // 25 harmonic pairs (one WMMA K=4 each)
constexpr int   CHUNK  = 2048;        // samples per block
constexpr int   NCHUNK = (T_LEN + CHUNK - 1) / CHUNK;   // 216
constexpr int   SUBW   = CHUNK / 8;   // samples per wave (256)
constexpr int   WITER  = SUBW / 16;   // 16 tiles of 16 samples per wave
constexpr float SCALE  = 2.0f / 44100.0f;               // N_BLADES / SAMPLE_RATE
constexpr float TWOPI  = 6.283185307179586f;

// sin/cos of (2*pi*frac) with frac in revolutions: v_sin_f32/v_cos_f32
// natively consume revolutions, so feed frac directly when possible.
__device__ __forceinline__ float sin_rev(float frac) {
#if __has_builtin(__builtin_amdgcn_sinf)
  return __builtin_amdgcn_sinf(frac);
#else
  return __sinf(TWOPI * frac);
#endif
}
__device__ __forceinline__ float cos_rev(float frac) {
#if __has_builtin(__builtin_amdgcn_cosf)
  return __builtin_amdgcn_cosf(frac);
#else
  return __cosf(TWOPI * frac);
#endif
}

// ---------------------------------------------------------------------------
// Pass 1: per-chunk sums of speed*SCALE (revolutions)
// ---------------------------------------------------------------------------
__global__ void pn_chunksum_kernel(const float* __restrict__ speed,
                                   float* __restrict__ chunkSums) {
  const int b     = blockIdx.y;
  const int chunk = blockIdx.x;
  const int tid   = threadIdx.x;
  const int lane  = tid & 31;
  const int wave  = tid >> 5;
  __shared__ float sW[8];

  const int  c0   = chunk * CHUNK;
  const int  base = b * T_LEN;
  float acc = 0.0f;
#pragma unroll
  for (int i = 0; i < CHUNK / 256; ++i) {
    int t = c0 + i * 256 + tid;
    if (t < T_LEN) acc += speed[base + t];
  }
  acc *= SCALE;
#pragma unroll
  for (int d = 16; d > 0; d >>= 1) acc += __shfl_down(acc, d, 32);
  if (lane == 0) sW[wave] = acc;
  __syncthreads();
  if (tid == 0) {
    float s = 0.0f;
#pragma unroll
    for (int w = 0; w < 8; ++w) s += sW[w];
    chunkSums[b * NCHUNK + chunk] = s;
  }
}

// ---------------------------------------------------------------------------
// Pass 2: exclusive scan of chunk sums (8 x 216, trivial) + per-(b,k) weights
//   w1 = exp(log_A)*cos(shift), w2 = exp(log_A)*sin(shift),
//   shift = phase_shift[b]*2*(k+1)
// ---------------------------------------------------------------------------
__global__ void pn_scan_weights_kernel(const float* __restrict__ phase_shift,
                                       const float* __restrict__ log_A,
                                       const float* __restrict__ chunkSums,
                                       float* __restrict__ chunkOff,
                                       float2* __restrict__ wPair) {
  const int tid = threadIdx.x;
  if (tid < NB * NH) {
    const int b = tid / NH;
    const int h = tid % NH;
    const float A     = expf(log_A[h]);
    const float shift = phase_shift[b] * 2.0f * (float)(h + 1);
    float2 w;
    w.x = A * cosf(shift);   // multiplies sin(k*phi)
    w.y = A * sinf(shift);   // multiplies cos(k*phi)
    wPair[b * NH + h] = w;
  }
  if (tid >= 448 && tid < 448 + NB) {
    const int b = tid - 448;
    float s = 0.0f;
    for (int c = 0; c < NCHUNK; ++c) {
      chunkOff[b * NCHUNK + c] = s;     // exclusive prefix in revolutions
      s += chunkSums[b * NCHUNK + c];
    }
  }
}

// ---------------------------------------------------------------------------
// Pass 3: synthesis. Block = (batch, chunk), 8 waves x 256 samples each.
// Per 16-sample tile: width-16 inclusive scan, one sincos of the
// fractional-revolution fundamental, then 25 x {4 FMA + 1
// v_wmma_f32_16x16x4_f32} with all operands in registers. Two interleaved
// accumulator chains keep the matrix pipe busy.
// ---------------------------------------------------------------------------
__global__ void __launch_bounds__(256)
pn_synth_kernel(const float* __restrict__ speed,
                const float* __restrict__ chunkOff,
                const float2* __restrict__ wPair,
                float* __restrict__ out) {
  const int b     = blockIdx.y;
  const int chunk = blockIdx.x;
  const int tid   = threadIdx.x;
  const int lane  = tid & 31;
  const int wave  = tid >> 5;
  const int hi    = lane >> 4;        // 0: odd harmonics, 1: even harmonics
  const int n     = lane & 15;        // owned sample index within tile

  __shared__ float  sWaveSum[8];
  __shared__ float2 sWgt[NH];

  if (tid < NH) sWgt[tid] = wPair[b * NH + tid];

  const int base  = b * T_LEN;
  const int wbase = chunk * CHUNK + wave * SUBW;

  // ---- phase A: per-wave sum of its 256 samples (full 32-lane efficiency) --
  float wsum = 0.0f;
#pragma unroll
  for (int i = 0; i < SUBW / 32; ++i) {
    int t = wbase + i * 32 + lane;
    float x = (t < T_LEN) ? speed[base + t] * SCALE : 0.0f;
    wsum += x;
  }
#pragma unroll
  for (int d = 16; d > 0; d >>= 1) wsum += __shfl_down(wsum, d, 32);
  if (lane == 0) sWaveSum[wave] = wsum;
  __syncthreads();

  float r = chunkOff[b * NCHUNK + chunk];   // exclusive prefix up to chunk
#pragma unroll
  for (int w = 0; w < 8; ++w) r += (w < wave) ? sWaveSum[w] : 0.0f;

  // ---- preload A-matrix operands into registers (invariant across tiles) --
  // Pair j, K-slots: K0=w1[2j] K1=w2[2j] (lo half), K2=w1[2j+1] K3=w2[2j+1]
  // (hi half)  ->  per-lane aW[j] = wPair[2j + hi]
  v2f aW[NPAIR];
#pragma unroll
  for (int j = 0; j < NPAIR; ++j) {
    const float2 wv = sWgt[2 * j + hi];
    aW[j].x = wv.x;
    aW[j].y = wv.y;
  }

  // ---- phase B: 16 tiles of 16 samples (both half-waves duplicate scan) ---
#pragma unroll 1
  for (int i = 0; i < WITER; ++i) {
    const int t = wbase + i * 16 + n;
    const bool valid = t < T_LEN;
    float x = valid ? speed[base + t] * SCALE : 0.0f;

    // width-16 inclusive scan (identical in both half-waves)
    float sx = x;
#pragma unroll
    for (int d = 1; d < 16; d <<= 1) {
      float v = __shfl_up(sx, (unsigned)d, 16);
      if (n >= d) sx += v;
    }
    const float total = __shfl(sx, 15, 16);
    const float rev   = r + sx;             // inclusive cumsum (revolutions)
    r += total;

    // fundamental: sincos straight from fractional revolutions
    const float frac = rev - floorf(rev);
    const float s1   = sin_rev(frac);       // v_sin_f32
    const float c1   = cos_rev(frac);       // v_cos_f32

    // double angle: rotation step for both parities, and hi-half start value
    const float s2k = 2.0f * s1 * c1;                 // sin 2phi
    const float c2k = fmaf(-2.0f * s1, s1, 1.0f);     // cos 2phi

    float s = hi ? s2k : s1;    // k = 2 (hi) or k = 1 (lo)
    float c = hi ? c2k : c1;

    v8f accP = {};   // even-pair chain
    v8f accQ = {};   // odd-pair chain
#pragma unroll
    for (int j = 0; j < NPAIR; j += 2) {
      {
        v2f bm;  bm.x = s;  bm.y = c;
        accP = __builtin_amdgcn_wmma_f32_16x16x4_f32(false, aW[j], false, bm,
                                                     (short)0, accP, false, false);
        const float ns = fmaf(s, c2k,  c * s2k);
        const float nc = fmaf(c, c2k, -s * s2k);
        s = ns;  c = nc;
      }
      if (j + 1 < NPAIR) {
        v2f bm;  bm.x = s;  bm.y = c;
        accQ = __builtin_amdgcn_wmma_f32_16x16x4_f32(false, aW[j + 1], false, bm,
                                                     (short)0, accQ, false, false);
        const float ns = fmaf(s, c2k,  c * s2k);
        const float nc = fmaf(c, c2k, -s * s2k);
        s = ns;  c = nc;
      }
    }

    // D rows are identical (A rows identical); D VGPR0 lane n = sample n.
    const float res = accP[0] + accQ[0];
    if (valid && hi == 0) out[base + t] = res;
  }
}

// ---------------------------------------------------------------------------
extern "C" void kernel_launch(void* const* d_in, const int* in_sizes, int n_in,
                              void* d_out, int out_size, void* d_ws, size_t ws_size,
                              hipStream_t stream) {
  const float* speed       = (const float*)d_in[0];   // [8, 441000]
  const float* phase_shift = (const float*)d_in[1];   // [8]
  const float* log_A       = (const float*)d_in[2];   // [50]
  float*       out         = (float*)d_out;           // [8, 441000]

  float*  ws        = (float*)d_ws;
  float*  chunkSums = ws;                               // NB*NCHUNK
  float*  chunkOff  = ws + NB * NCHUNK;                 // NB*NCHUNK
  float2* wPair     = (float2*)(ws + 2 * NB * NCHUNK);  // NB*NH float2 (8B aligned)

  pn_chunksum_kernel<<<dim3(NCHUNK, NB), 256, 0, stream>>>(speed, chunkSums);
  pn_scan_weights_kernel<<<1, 512, 0, stream>>>(phase_shift, log_A,
                                                chunkSums, chunkOff, wPair);
  pn_synth_kernel<<<dim3(NCHUNK, NB), 256, 0, stream>>>(speed, chunkOff, wPair, out);
}